// GNNHeteroPooling_71983651881218
// MI455X (gfx1250) — compile-verified
//
#include <hip/hip_runtime.h>

#define H 128
#define TM 16

typedef __attribute__((ext_vector_type(2))) float v2f;
typedef __attribute__((ext_vector_type(8))) float v8f;

// ---------------- zero fill ----------------
__global__ void zero_f4(float4* __restrict__ p, size_t n4) {
  size_t i = (size_t)blockIdx.x * blockDim.x + threadIdx.x;
  size_t stride = (size_t)gridDim.x * blockDim.x;
  float4 z = make_float4(0.f, 0.f, 0.f, 0.f);
  for (; i < n4; i += stride) p[i] = z;
}

__global__ void zero_u32(unsigned* __restrict__ p, int n) {
  int i = blockIdx.x * blockDim.x + threadIdx.x;
  if (i < n) p[i] = 0u;
}

// ---------------- edge scatter-add: one wave per edge, float4 per lane ----------------
__global__ void scatter_add(const float* __restrict__ xsrc, const int* __restrict__ edge,
                            float* __restrict__ agg, int E_) {
  int gt = blockIdx.x * blockDim.x + threadIdx.x;
  int e = gt >> 5;
  int lane = gt & 31;
  if (e >= E_) return;
  int src = edge[e];        // edge[0][e]
  int dst = edge[E_ + e];   // edge[1][e]
  float4 v = ((const float4*)(xsrc + (size_t)src * H))[lane];
  float* a = agg + (size_t)dst * H + lane * 4;
  atomicAdd(a + 0, v.x);
  atomicAdd(a + 1, v.y);
  atomicAdd(a + 2, v.z);
  atomicAdd(a + 3, v.w);
}

// ---------------- fused dual-GEMM: Out = Agg@Wrel + brel + X@Wroot (opt ReLU) -----------
// 8 waves/block, wave w owns output columns [16w,16w+16). WMMA f32 16x16x4, K=128.
// B operands (weight slabs) live in VGPRs: wave-invariant 16-column slab, 64 v2f total.
// LDS holds only the two 16x128 A tiles, packed for per-lane v2f WMMA feeds:
//   v2f ap[r*16 + m] = (A[m][2r], A[m][2r+1]), r<64, m<16
// => half-wave c=0 hits banks [0..31], c=1 hits [32..63]: conflict-free ds_load_b64.
__global__ __launch_bounds__(256)
void gemm_fused(const float* __restrict__ Agg, const float* __restrict__ X,
                const float* __restrict__ Wrel, const float* __restrict__ brel,
                const float* __restrict__ Wroot, float* __restrict__ Out,
                int mtiles, int relu) {
  __shared__ float smem[4096];  // 16 KB: agg tile [0..2048), x tile [2048..4096)
  v2f* aagg_p = (v2f*)(smem);
  v2f* ax_p   = (v2f*)(smem + 2048);

  int tid  = threadIdx.x;
  int lane = tid & 31;
  int wave = tid >> 5;
  int c  = lane >> 4;   // half-wave: selects K offset (A) / K row-pair (B)
  int nl = lane & 15;   // M row (for A) / N col (for B,C)
  int nb = wave * 16;   // column slab base for this wave
  float bias = brel[nb + nl];

  // Preload this wave's B slabs into registers: wr[k4] = (W[4k4+2c][n], W[4k4+2c+1][n]).
  v2f wr[32], wo[32];
#pragma unroll
  for (int k4 = 0; k4 < 32; ++k4) {
    int r2 = (2 * k4 + c) * 2;  // = 4*k4 + 2*c
    int col = nb + nl;
    v2f a = { Wrel[(size_t)r2 * H + col],  Wrel[(size_t)(r2 + 1) * H + col] };
    wr[k4] = a;
    v2f b = { Wroot[(size_t)r2 * H + col], Wroot[(size_t)(r2 + 1) * H + col] };
    wo[k4] = b;
  }

  for (int mt = blockIdx.x; mt < mtiles; mt += gridDim.x) {
    int m_base = mt * TM;
    __syncthreads();  // previous iteration finished reading A tiles
    // Stage 16-row agg and x tiles, repacked for per-lane v2f WMMA feeds.
    for (int i = tid; i < TM * 32; i += 256) {
      int m = i >> 5, j = i & 31;
      float4 v = ((const float4*)(Agg + (size_t)(m_base + m) * H))[j];
      v2f v01 = { v.x, v.y }; v2f v23 = { v.z, v.w };
      aagg_p[(2 * j) * 16 + m]     = v01;
      aagg_p[(2 * j + 1) * 16 + m] = v23;
      float4 u = ((const float4*)(X + (size_t)(m_base + m) * H))[j];
      v2f u01 = { u.x, u.y }; v2f u23 = { u.z, u.w };
      ax_p[(2 * j) * 16 + m]     = u01;
      ax_p[(2 * j + 1) * 16 + m] = u23;
    }
    __syncthreads();

    v8f acc = {};
#pragma unroll
    for (int k4 = 0; k4 < 32; ++k4) {
      int r = 2 * k4 + c;
      v2f aA = aagg_p[r * 16 + nl];
      acc = __builtin_amdgcn_wmma_f32_16x16x4_f32(false, aA, false, wr[k4],
                                                  (short)0, acc, false, false);
      v2f aX = ax_p[r * 16 + nl];
      acc = __builtin_amdgcn_wmma_f32_16x16x4_f32(false, aX, false, wo[k4],
                                                  (short)0, acc, false, false);
    }

    // C layout: VGPR r of lane -> row m_base + c*8 + r, col nb + nl.
    float* op = Out + (size_t)(m_base + c * 8) * H + nb + nl;
#pragma unroll
    for (int r = 0; r < 8; ++r) {
      float v = acc[r] + bias;
      if (relu) v = v > 0.f ? v : 0.f;
      op[(size_t)r * H] = v;
    }
  }
}

// ---------------- segment-max pooling via monotone-uint atomicMax ----------------
__device__ __forceinline__ unsigned fmap(float f) {
  unsigned u = __float_as_uint(f);
  return (u & 0x80000000u) ? ~u : (u | 0x80000000u);
}
__device__ __forceinline__ float funmap(unsigned u) {
  u = (u & 0x80000000u) ? (u & 0x7FFFFFFFu) : ~u;
  return __uint_as_float(u);
}

__global__ void pool_max(const float* __restrict__ x, const int* __restrict__ batch,
                         unsigned* __restrict__ pooled, int Nn) {
  int gt = blockIdx.x * blockDim.x + threadIdx.x;
  int node = gt >> 5, lane = gt & 31;
  if (node >= Nn) return;
  int g = batch[node];
  float4 v = ((const float4*)(x + (size_t)node * H))[lane];
  unsigned* p = pooled + (size_t)g * H + lane * 4;
  atomicMax(p + 0, fmap(v.x));
  atomicMax(p + 1, fmap(v.y));
  atomicMax(p + 2, fmap(v.z));
  atomicMax(p + 3, fmap(v.w));
}

// ---------------- head MLPs + final combine (G graphs, trivial) ----------------
__global__ void head_mlp(const unsigned* __restrict__ pa, const unsigned* __restrict__ pb,
                         const float* __restrict__ w1a, const float* __restrict__ b1a,
                         const float* __restrict__ w2a, const float* __restrict__ b2a,
                         const float* __restrict__ w1b, const float* __restrict__ b1b,
                         const float* __restrict__ w2b, const float* __restrict__ b2b,
                         const float* __restrict__ ow, const float* __restrict__ obias,
                         float* __restrict__ out, int G_) {
  int g = blockIdx.x * blockDim.x + threadIdx.x;
  if (g >= G_) return;
  float ha[5], hb[5];
#pragma unroll
  for (int j = 0; j < 5; ++j) { ha[j] = b1a[j]; hb[j] = b1b[j]; }
  for (int k = 0; k < H; ++k) {
    float xa = funmap(pa[(size_t)g * H + k]);
    float xb = funmap(pb[(size_t)g * H + k]);
#pragma unroll
    for (int j = 0; j < 5; ++j) {
      ha[j] += xa * w1a[k * 5 + j];
      hb[j] += xb * w1b[k * 5 + j];
    }
  }
  float oa = b2a[0], obv = b2b[0];
#pragma unroll
  for (int j = 0; j < 5; ++j) {
    oa  += (ha[j] > 0.f ? ha[j] : 0.f) * w2a[j];
    obv += (hb[j] > 0.f ? hb[j] : 0.f) * w2b[j];
  }
  out[g] = oa * ow[0] + obv * ow[1] + obias[0];
}

// ---------------- orchestration ----------------
extern "C" void kernel_launch(void* const* d_in, const int* in_sizes, int n_in,
                              void* d_out, int out_size, void* d_ws, size_t ws_size,
                              hipStream_t stream) {
  (void)n_in; (void)ws_size;
  const float* x_a    = (const float*)d_in[0];
  const float* x_b    = (const float*)d_in[1];
  const float* W_rel  = (const float*)d_in[2];
  const float* b_rel  = (const float*)d_in[3];
  const float* W_root = (const float*)d_in[4];
  const float* w1a = (const float*)d_in[5];
  const float* b1a = (const float*)d_in[6];
  const float* w2a = (const float*)d_in[7];
  const float* b2a = (const float*)d_in[8];
  const float* w1b = (const float*)d_in[9];
  const float* b1b = (const float*)d_in[10];
  const float* w2b = (const float*)d_in[11];
  const float* b2b = (const float*)d_in[12];
  const float* ow  = (const float*)d_in[13];
  const float* ob  = (const float*)d_in[14];
  const int* edge_ab = (const int*)d_in[15];
  const int* edge_ba = (const int*)d_in[16];
  const int* batch_a = (const int*)d_in[17];
  const int* batch_b = (const int*)d_in[18];

  const int Nn = in_sizes[0] / H;        // 100000
  const int E_ = in_sizes[15] / 2;       // 640000
  const int G_ = out_size;               // 64
  const int L_ = in_sizes[3] / (2 * H);  // 3

  size_t nh = (size_t)Nn * H;
  float* ws = (float*)d_ws;
  float* agg_a = ws;                 // agg_a, agg_b contiguous (zeroed together)
  float* agg_b = ws + nh;
  float* xa0 = ws + 2 * nh;
  float* xb0 = ws + 3 * nh;
  float* xa1 = ws + 4 * nh;
  float* xb1 = ws + 5 * nh;
  unsigned* pa = (unsigned*)(ws + 6 * nh);
  unsigned* pb = pa + (size_t)G_ * H;

  const int mtiles = (Nn + TM - 1) / TM;
  const int sc_blocks = (E_ * 32 + 255) / 256;

  const float* xca = x_a;
  const float* xcb = x_b;
  float* bufs[2][2] = {{xa0, xb0}, {xa1, xb1}};

  for (int l = 0; l < L_; ++l) {
    zero_f4<<<4096, 256, 0, stream>>>((float4*)agg_a, 2 * nh / 4);
    scatter_add<<<sc_blocks, 256, 0, stream>>>(xcb, edge_ba, agg_a, E_);
    scatter_add<<<sc_blocks, 256, 0, stream>>>(xca, edge_ab, agg_b, E_);
    int relu = (l < L_ - 1) ? 1 : 0;
    float* na = bufs[l & 1][0];
    float* nb = bufs[l & 1][1];
    // dst type A uses rel/root index [l,1]; dst type B uses [l,0]
    gemm_fused<<<640, 256, 0, stream>>>(
        agg_a, xca, W_rel + (size_t)(l * 2 + 1) * H * H, b_rel + (size_t)(l * 2 + 1) * H,
        W_root + (size_t)(l * 2 + 1) * H * H, na, mtiles, relu);
    gemm_fused<<<640, 256, 0, stream>>>(
        agg_b, xcb, W_rel + (size_t)(l * 2 + 0) * H * H, b_rel + (size_t)(l * 2 + 0) * H,
        W_root + (size_t)(l * 2 + 0) * H * H, nb, mtiles, relu);
    xca = na; xcb = nb;
  }

  zero_u32<<<(2 * G_ * H + 255) / 256, 256, 0, stream>>>(pa, 2 * G_ * H);
  pool_max<<<(Nn * 32 + 255) / 256, 256, 0, stream>>>(xca, batch_a, pa, Nn);
  pool_max<<<(Nn * 32 + 255) / 256, 256, 0, stream>>>(xcb, batch_b, pb, Nn);

  head_mlp<<<(G_ + 63) / 64, 64, 0, stream>>>(pa, pb, w1a, b1a, w2a, b2a,
                                              w1b, b1b, w2b, b2b, ow, ob,
                                              (float*)d_out, G_);
}